// HybridSECBAM_81707457839700
// MI455X (gfx1250) — compile-verified
//
#include <hip/hip_runtime.h>

// ---------------------------------------------------------------------------
// HybridSECBAM on MI455X (gfx1250, wave32).
// Memory-bound: ~0.93 GB min traffic -> ~40us @ 23.3 TB/s. GEMMs are tiny and
// run on f32 WMMA (V_WMMA_F32_16X16X4_F32) to use the matrix pipe at full fp32
// precision. 5 kernels, all stream-ordered, deterministic (no float atomics).
// N-padding for the narrow layers is done with clamped in-bounds addresses +
// value selects (v_cndmask) so no EXEC-divergent predicated loads appear in
// the WMMA inner loops.
// ---------------------------------------------------------------------------

#define Bz 32
#define Hz 112
#define Wz 112
#define Cz 192

typedef __attribute__((ext_vector_type(2))) float v2f;
typedef __attribute__((ext_vector_type(8))) float v8f;

__device__ __forceinline__ float sigmoidf_(float v) {
  return 1.0f / (1.0f + __expf(-v));
}

// ---------------------------------------------------------------------------
// Kernel 1: fused pooling. One block per (b,h) row (192 threads = 6 waves,
// thread == channel). Single read of x produces:
//   - per-row channel partial sum/max  [B*H][C]   (reduced over h in kernel 2)
//   - spatial avg/max over C           [B][H][W]
// ---------------------------------------------------------------------------
__global__ __launch_bounds__(192) void pool_kernel(
    const float* __restrict__ x,
    float* __restrict__ p_sum, float* __restrict__ p_max,
    float* __restrict__ avgsp, float* __restrict__ maxsp) {
  const int row = blockIdx.x;              // b*H + h
  const int c = threadIdx.x;               // channel
  const int lane = c & 31, wv = c >> 5;
  __shared__ float sp_s[Wz][6];
  __shared__ float sp_m[Wz][6];

  const float* px = x + (size_t)row * Wz * Cz + c;
  float csum = 0.0f, cmax = -3.402823466e+38f;
  for (int w = 0; w < Wz; ++w) {
    float v = px[(size_t)w * Cz];
    csum += v;
    cmax = fmaxf(cmax, v);
    // wave32 reduction of this pixel's 32 channels handled by this wave
    float s = v, m = v;
    for (int off = 16; off > 0; off >>= 1) {
      s += __shfl_xor(s, off, 32);
      m = fmaxf(m, __shfl_xor(m, off, 32));
    }
    if (lane == 0) { sp_s[w][wv] = s; sp_m[w][wv] = m; }
  }
  p_sum[(size_t)row * Cz + c] = csum;
  p_max[(size_t)row * Cz + c] = cmax;
  __syncthreads();
  if (c < Wz) {
    float s = 0.0f, m = -3.402823466e+38f;
    for (int j = 0; j < 6; ++j) { s += sp_s[c][j]; m = fmaxf(m, sp_m[c][j]); }
    avgsp[(size_t)row * Wz + c] = s * (1.0f / (float)Cz);
    maxsp[(size_t)row * Wz + c] = m;
  }
}

// ---------------------------------------------------------------------------
// Kernel 2: reduce per-row channel partials over H -> avg_pool/max_pool [B][C]
// ---------------------------------------------------------------------------
__global__ __launch_bounds__(256) void chan_reduce_kernel(
    const float* __restrict__ p_sum, const float* __restrict__ p_max,
    float* __restrict__ avgp, float* __restrict__ maxp) {
  const int tid = blockIdx.x * 256 + threadIdx.x;  // 0..B*C-1 (=6144)
  if (tid >= Bz * Cz) return;
  const int b = tid / Cz, c = tid % Cz;
  const float* ps = p_sum + (size_t)b * Hz * Cz + c;
  const float* pm = p_max + (size_t)b * Hz * Cz + c;
  float s = 0.0f, m = -3.402823466e+38f;
  for (int h = 0; h < Hz; ++h) {
    s += ps[(size_t)h * Cz];
    m = fmaxf(m, pm[(size_t)h * Cz]);
  }
  avgp[tid] = s * (1.0f / (float)(Hz * Wz));
  maxp[tid] = m;
}

// ---------------------------------------------------------------------------
// WMMA helpers (V_WMMA_F32_16X16X4_F32, wave32).
// ISA layouts (CDNA5 ISA):
//   A 16x4 f32 : lane&15 = M, half=lane>>4, VGPR v -> K = 2*half + v
//   B 4x16 f32 : lane&15 = N, half=lane>>4, VGPR v -> K = 2*half + v
//   C/D 16x16  : lane&15 = N, VGPR v -> M = v + 8*half
// ---------------------------------------------------------------------------

// Layer-1 tile: sOut[16x16 tile] = swish(A @ W + bias); N columns >= Nvalid
// are zero-padded via clamped-address loads + value selects (keeps loads
// unconditional -> no EXEC-divergent branches in the WMMA loop).
__device__ __forceinline__ void gemm_tile_l1(
    const float* __restrict__ sA, int lda, int K,
    const float* __restrict__ W, int ldw, int Nvalid,
    const float* __restrict__ bias,
    int mtile, int ntile,
    float* __restrict__ sOut, int ldo) {
  const int lane = threadIdx.x & 31;
  const int half = lane >> 4;
  const int l15 = lane & 15;
  const int m0 = mtile * 16;
  const int n = ntile * 16 + l15;
  const bool nok = (n < Nvalid);
  const int nc = nok ? n : 0;  // clamped, always in-bounds
  v8f c = {};
  for (int k0 = 0; k0 < K; k0 += 4) {
    const int ka = k0 + 2 * half;
    v2f a, b;
    a.x = sA[(m0 + l15) * lda + ka];
    a.y = sA[(m0 + l15) * lda + ka + 1];
    const float w0 = W[(size_t)ka * ldw + nc];        // unconditional load
    const float w1 = W[(size_t)(ka + 1) * ldw + nc];  // unconditional load
    b.x = nok ? w0 : 0.0f;  // v_cndmask on value, not on load
    b.y = nok ? w1 : 0.0f;
    c = __builtin_amdgcn_wmma_f32_16x16x4_f32(false, a, false, b, (short)0, c,
                                              false, false);
  }
  const float bv = bias[nc];
  const float bn = nok ? bv : 0.0f;
#pragma unroll
  for (int v = 0; v < 8; ++v) {
    const int m = m0 + v + 8 * half;
    float h = c[v] + bn;
    h = h * sigmoidf_(h);  // swish
    sOut[m * ldo + ntile * 16 + l15] = nok ? h : 0.0f;
  }
}

// Layer-2 accumulate: c += A(LDS) @ W(global, ldw=192), exact K (no padding,
// all N columns valid -> fully unconditional loads).
__device__ __forceinline__ void gemm_tile_acc(
    v8f& c, const float* __restrict__ sA, int lda, int K,
    const float* __restrict__ W, int ldw, int mtile, int ntile) {
  const int lane = threadIdx.x & 31;
  const int half = lane >> 4;
  const int l15 = lane & 15;
  const int m0 = mtile * 16;
  const int n = ntile * 16 + l15;
  for (int k0 = 0; k0 < K; k0 += 4) {
    const int ka = k0 + 2 * half;
    v2f a, b;
    a.x = sA[(m0 + l15) * lda + ka];
    a.y = sA[(m0 + l15) * lda + ka + 1];
    b.x = W[(size_t)ka * ldw + n];
    b.y = W[(size_t)(ka + 1) * ldw + n];
    c = __builtin_amdgcn_wmma_f32_16x16x4_f32(false, a, false, b, (short)0, c,
                                              false, false);
  }
}

// ---------------------------------------------------------------------------
// Kernel 3: SE + CBAM channel attention via WMMA. One block, 8 waves; each
// wave owns whole 16x16 output tiles (EXEC all-ones around WMMA, branches are
// wave-uniform).
//   se  = sigmoid(swish(avg@se_w1+b1)@se_w2+b2)               [32][192]
//   ca  = sigmoid(mlp(avg)+mlp(max)) with mlp(v)=swish(v@w1+b1)@w2+b2
// ---------------------------------------------------------------------------
__global__ __launch_bounds__(256) void attn_vec_kernel(
    const float* __restrict__ avgp, const float* __restrict__ maxp,
    const float* __restrict__ se_w1, const float* __restrict__ se_b1,
    const float* __restrict__ se_w2, const float* __restrict__ se_b2,
    const float* __restrict__ mlp_w1, const float* __restrict__ mlp_b1,
    const float* __restrict__ mlp_w2, const float* __restrict__ mlp_b2,
    float* __restrict__ se_out, float* __restrict__ ca_out) {
  __shared__ float sAvg[Bz * Cz];   // 32x192
  __shared__ float sMax[Bz * Cz];   // 32x192
  __shared__ float sHse[Bz * 16];   // 32x12 (padded to 16)
  __shared__ float sHavg[Bz * 32];  // 32x24 (padded to 32)
  __shared__ float sHmax[Bz * 32];

  const int tid = threadIdx.x;
  for (int i = tid; i < Bz * Cz; i += 256) {
    sAvg[i] = avgp[i];
    sMax[i] = maxp[i];
  }
  __syncthreads();

  const int wave = tid >> 5;
  // Phase A: hidden layers, 10 tiles (se:2, mlp(avg):4, mlp(max):4), K=192.
  for (int t = wave; t < 10; t += 8) {
    if (t < 2) {
      gemm_tile_l1(sAvg, Cz, Cz, se_w1, 12, 12, se_b1, t, 0, sHse, 16);
    } else if (t < 6) {
      const int u = t - 2;
      gemm_tile_l1(sAvg, Cz, Cz, mlp_w1, 24, 24, mlp_b1, u & 1, u >> 1, sHavg, 32);
    } else {
      const int u = t - 6;
      gemm_tile_l1(sMax, Cz, Cz, mlp_w1, 24, 24, mlp_b1, u & 1, u >> 1, sHmax, 32);
    }
  }
  __syncthreads();

  // Phase B: output layers, 48 tiles (2 M-tiles x 12 N-tiles, se + channel).
  const int lane = tid & 31;
  const int half = lane >> 4;
  const int l15 = lane & 15;
  for (int t = wave; t < 48; t += 8) {
    v8f c = {};
    if (t < 24) {
      const int mtile = t / 12, ntile = t % 12;
      gemm_tile_acc(c, sHse, 16, 12, se_w2, Cz, mtile, ntile);
      const int n = ntile * 16 + l15;
      const float bn = se_b2[n];
#pragma unroll
      for (int v = 0; v < 8; ++v) {
        const int m = mtile * 16 + v + 8 * half;
        se_out[m * Cz + n] = sigmoidf_(c[v] + bn);
      }
    } else {
      const int u = t - 24;
      const int mtile = u / 12, ntile = u % 12;
      gemm_tile_acc(c, sHavg, 32, 24, mlp_w2, Cz, mtile, ntile);
      gemm_tile_acc(c, sHmax, 32, 24, mlp_w2, Cz, mtile, ntile);
      const int n = ntile * 16 + l15;
      const float bn = 2.0f * mlp_b2[n];  // mlp bias added once per branch
#pragma unroll
      for (int v = 0; v < 8; ++v) {
        const int m = mtile * 16 + v + 8 * half;
        ca_out[m * Cz + n] = sigmoidf_(c[v] + bn);
      }
    }
  }
}

// ---------------------------------------------------------------------------
// Kernel 4: 7x7x2->1 SAME conv on (avg_sp,max_sp) + sigmoid -> spatial_att.
// conv_k layout HWIO: k[(kh*7+kw)*2 + i]. 39 MFLOP total -> trivial.
// ---------------------------------------------------------------------------
__global__ __launch_bounds__(256) void spatial_conv_kernel(
    const float* __restrict__ avgsp, const float* __restrict__ maxsp,
    const float* __restrict__ conv_k, const float* __restrict__ conv_b,
    float* __restrict__ satt) {
  __shared__ float sk[98];
  __shared__ float sb;
  const int tid = threadIdx.x;
  if (tid < 98) sk[tid] = conv_k[tid];
  if (tid == 0) sb = conv_b[0];
  __syncthreads();

  const int p = blockIdx.x * 256 + tid;  // 0..B*H*W-1 (exact multiple)
  const int b = p / (Hz * Wz);
  const int hw = p % (Hz * Wz);
  const int h = hw / Wz, w = hw % Wz;
  const float* pa = avgsp + (size_t)b * Hz * Wz;
  const float* pm = maxsp + (size_t)b * Hz * Wz;
  float acc = sb;
#pragma unroll
  for (int kh = 0; kh < 7; ++kh) {
    const int ih = h + kh - 3;
    if (ih < 0 || ih >= Hz) continue;
#pragma unroll
    for (int kw = 0; kw < 7; ++kw) {
      const int iw = w + kw - 3;
      if (iw < 0 || iw >= Wz) continue;
      const int ki = (kh * 7 + kw) * 2;
      const float va = pa[ih * Wz + iw];
      const float vm = pm[ih * Wz + iw];
      acc = fmaf(va, sk[ki], acc);
      acc = fmaf(vm, sk[ki + 1], acc);
    }
  }
  satt[p] = sigmoidf_(acc);
}

// ---------------------------------------------------------------------------
// Kernel 5: out = x * se[b,c] * ca[b,c] * satt[b,h,w], float4-vectorized
// (C=192 -> 48 float4 per pixel). Bandwidth-bound: 308MB rd + 308MB wr.
// ---------------------------------------------------------------------------
__global__ __launch_bounds__(256) void fuse_kernel(
    const float* __restrict__ x, const float* __restrict__ se,
    const float* __restrict__ ca, const float* __restrict__ satt,
    float* __restrict__ out) {
  const long long i4 = (long long)blockIdx.x * 256 + threadIdx.x;
  const int c4 = (int)(i4 % 48);
  const long long pix = i4 / 48;
  const int b = (int)(pix / (Hz * Wz));
  const int c = c4 * 4;
  const float4 xv = reinterpret_cast<const float4*>(x)[i4];
  const float sp = satt[pix];
  const float* g1 = se + b * Cz + c;
  const float* g2 = ca + b * Cz + c;
  float4 o;
  o.x = xv.x * (g1[0] * g2[0] * sp);
  o.y = xv.y * (g1[1] * g2[1] * sp);
  o.z = xv.z * (g1[2] * g2[2] * sp);
  o.w = xv.w * (g1[3] * g2[3] * sp);
  reinterpret_cast<float4*>(out)[i4] = o;
}

// ---------------------------------------------------------------------------
extern "C" void kernel_launch(void* const* d_in, const int* in_sizes, int n_in,
                              void* d_out, int out_size, void* d_ws,
                              size_t ws_size, hipStream_t stream) {
  const float* x      = (const float*)d_in[0];
  const float* se_w1  = (const float*)d_in[1];
  const float* se_b1  = (const float*)d_in[2];
  const float* se_w2  = (const float*)d_in[3];
  const float* se_b2  = (const float*)d_in[4];
  const float* mlp_w1 = (const float*)d_in[5];
  const float* mlp_b1 = (const float*)d_in[6];
  const float* mlp_w2 = (const float*)d_in[7];
  const float* mlp_b2 = (const float*)d_in[8];
  const float* conv_k = (const float*)d_in[9];
  const float* conv_b = (const float*)d_in[10];
  float* out = (float*)d_out;
  float* ws = (float*)d_ws;

  // Workspace layout (floats), total ~10.4 MB.
  float* p_sum = ws;                         // [B*H][C]  = 688128
  float* p_max = p_sum + Bz * Hz * Cz;       // 688128
  float* avgp  = p_max + Bz * Hz * Cz;       // [B][C]    = 6144
  float* maxp  = avgp + Bz * Cz;             // 6144
  float* avgsp = maxp + Bz * Cz;             // [B][H][W] = 401408
  float* maxsp = avgsp + Bz * Hz * Wz;       // 401408
  float* se    = maxsp + Bz * Hz * Wz;       // [B][C]    = 6144
  float* ca    = se + Bz * Cz;               // 6144
  float* satt  = ca + Bz * Cz;               // [B][H][W] = 401408

  pool_kernel<<<Bz * Hz, 192, 0, stream>>>(x, p_sum, p_max, avgsp, maxsp);
  chan_reduce_kernel<<<(Bz * Cz + 255) / 256, 256, 0, stream>>>(p_sum, p_max,
                                                               avgp, maxp);
  attn_vec_kernel<<<1, 256, 0, stream>>>(avgp, maxp, se_w1, se_b1, se_w2,
                                         se_b2, mlp_w1, mlp_b1, mlp_w2, mlp_b2,
                                         se, ca);
  spatial_conv_kernel<<<(Bz * Hz * Wz) / 256, 256, 0, stream>>>(avgsp, maxsp,
                                                               conv_k, conv_b,
                                                               satt);
  fuse_kernel<<<(long long)(Bz * Hz * Wz) * Cz / 4 / 256, 256, 0, stream>>>(
      x, se, ca, satt, out);
}